// CriterionLP_intra_84447646974570
// MI455X (gfx1250) — compile-verified
//
#include <hip/hip_runtime.h>

// Problem constants (match reference setup_inputs()).
constexpr int Cdim  = 2048;   // feature dim (K)
constexpr int Bq    = 2048;   // query rows (B)
constexpr int Ncols = 16384;  // Bs * TOPK support columns (N)

constexpr int KC = 16;         // K per LDS stage (double-buffered)
constexpr int SB = 20;         // padded LDS row stride in floats (80B: 16B-aligned, bank-conflict-free)
constexpr int TN = 256;        // columns per block
constexpr int CH = Ncols / TN; // 64 column chunks
constexpr int NW = 4;          // waves per block (128 threads)
constexpr int NT = 4;          // 16-col tiles per wave -> 4 independent WMMA chains

typedef float v2f __attribute__((ext_vector_type(2)));
typedef float v8f __attribute__((ext_vector_type(8)));
typedef int   v4i __attribute__((ext_vector_type(4)));

#if defined(__AMDGCN__) && __has_builtin(__builtin_amdgcn_global_load_async_to_lds_b128)
#define USE_ASYNC_LDS 1
typedef __attribute__((address_space(1))) v4i* gv4i_p;  // global-AS v4i*
typedef __attribute__((address_space(3))) v4i* lv4i_p;  // LDS-AS v4i*
#else
#define USE_ASYNC_LDS 0
#endif

// Copy 16B global -> LDS. Async DMA (ASYNCcnt) on gfx1250, sync fallback otherwise.
__device__ __forceinline__ void copy16(const float* __restrict__ g, float* l) {
#if USE_ASYNC_LDS
    __builtin_amdgcn_global_load_async_to_lds_b128(
        (gv4i_p)(unsigned long long)g, (lv4i_p)l, /*imm offset*/0, /*cpol*/0);
#else
    *(float4*)l = *(const float4*)g;
#endif
}

__device__ __forceinline__ void wait_async_done() {
#if USE_ASYNC_LDS
#if __has_builtin(__builtin_amdgcn_s_wait_asynccnt)
    __builtin_amdgcn_s_wait_asynccnt(0);
#else
    asm volatile("s_wait_asynccnt 0" ::: "memory");
#endif
#endif
}

__global__ __launch_bounds__(128) void lp_gemm_reduce(
    const float* __restrict__ feats,      // [Bq, Cdim]
    const float* __restrict__ fs,         // [Ncols, Cdim]
    const int*   __restrict__ labels,     // [Bq]
    const int*   __restrict__ labels_s,   // [Ncols]
    float* __restrict__ wsMin,            // [Bq, CH]
    float* __restrict__ wsSum)            // [Bq, CH]
{
    __shared__ float As[2][16 * SB];      // 16 rows  x KC, double-buffered
    __shared__ float Bs[2][TN * SB];      // 256 cols x KC, double-buffered
    __shared__ float sMin[NW][16];
    __shared__ float sSum[NW][16];
    __shared__ int   sRowLab[16];

    const int tid  = threadIdx.x;
    const int wv   = tid >> 5;            // wave 0..3
    const int ln   = tid & 31;
    const int half = ln >> 4;             // lane half: selects K pair per f32 fragment layout
    const int l16  = ln & 15;
    const int koff = half * 2;

    const int rb = blockIdx.y * 16;       // row base
    const int cb = blockIdx.x * TN;       // col base

    if (tid < 16) sRowLab[tid] = labels[rb + tid];

    // Column labels for this wave's four 16-col tiles.
    int cl[NT];
    #pragma unroll
    for (int t = 0; t < NT; ++t)
        cl[t] = labels_s[cb + wv * 64 + t * 16 + l16];

    v8f acc[NT] = {};

    // Stage one K-tile (16 rows x 16 k of A, 256 cols x 16 k of B) into LDS buffer `buf`.
    auto stage = [&](int buf, int k0) {
        if (tid < 64) {                                   // A: 64 x b128 chunks
            const int row = tid >> 2;
            const int kq  = (tid & 3) * 4;
            copy16(&feats[(size_t)(rb + row) * Cdim + k0 + kq],
                   &As[buf][row * SB + kq]);
        }
        #pragma unroll
        for (int i = 0; i < 8; ++i) {                     // B: 1024 x b128 chunks, 8/thread
            const int j  = tid + i * 128;
            const int c  = j >> 2;
            const int kq = (j & 3) * 4;
            copy16(&fs[(size_t)(cb + c) * Cdim + k0 + kq],
                   &Bs[buf][c * SB + kq]);
        }
    };

    // Prologue: fill buffer 0.
    stage(0, 0);
    wait_async_done();
    __syncthreads();

    int buf = 0;
    for (int k0 = 0; k0 < Cdim; k0 += KC) {
        // Prefetch next K-tile into the other buffer while computing this one.
        if (k0 + KC < Cdim) stage(buf ^ 1, k0 + KC);

        // Preload all fragments for this K-tile, then run 16 WMMAs in 4 chains.
        v2f a[4];
        #pragma unroll
        for (int s = 0; s < 4; ++s)
            a[s] = *(const v2f*)&As[buf][l16 * SB + s * 4 + koff];
        v2f b[NT][4];
        #pragma unroll
        for (int t = 0; t < NT; ++t)
            #pragma unroll
            for (int s = 0; s < 4; ++s)
                b[t][s] = *(const v2f*)&Bs[buf][(wv * 64 + t * 16 + l16) * SB + s * 4 + koff];
        #pragma unroll
        for (int s = 0; s < 4; ++s)
            #pragma unroll
            for (int t = 0; t < NT; ++t)
                acc[t] = __builtin_amdgcn_wmma_f32_16x16x4_f32(
                    false, a[s], false, b[t][s], (short)0, acc[t], false, false);

        wait_async_done();   // our async writes into buf^1 have landed
        __syncthreads();     // everyone's writes visible; safe to swap
        buf ^= 1;
    }

    // Fused epilogue: E = exp(sim/0.05); per-row hardest-positive min + negative sum.
    // C/D layout: VGPR g, lanes 0-15 -> row g, lanes 16-31 -> row g+8; col = l16 + 16*t.
    float pm[8], ns[8];
    #pragma unroll
    for (int g = 0; g < 8; ++g) {
        const int rl = sRowLab[g + half * 8];
        pm[g] = __builtin_inff();
        ns[g] = 0.0f;
        #pragma unroll
        for (int t = 0; t < NT; ++t) {
            const float e = __expf(acc[t][g] * 20.0f);
            const bool  p = (rl == cl[t]);
            pm[g] = fminf(pm[g], p ? e : __builtin_inff());
            ns[g] += p ? 0.0f : e;
        }
    }
    // Reduce across the 16 lanes of each half-wave (xor masks < 16 stay in half).
    #pragma unroll
    for (int off = 1; off < 16; off <<= 1) {
        #pragma unroll
        for (int g = 0; g < 8; ++g) {
            pm[g] = fminf(pm[g], __shfl_xor(pm[g], off, 32));
            ns[g] += __shfl_xor(ns[g], off, 32);
        }
    }
    if (l16 == 0) {
        #pragma unroll
        for (int g = 0; g < 8; ++g) {
            sMin[wv][g + half * 8] = pm[g];
            sSum[wv][g + half * 8] = ns[g];
        }
    }
    __syncthreads();
    // Combine the waves; write deterministic per-(row, chunk) partials.
    if (tid < 16) {
        float mn = __builtin_inff();
        float sm = 0.0f;
        #pragma unroll
        for (int w = 0; w < NW; ++w) {
            mn  = fminf(mn, sMin[w][tid]);
            sm += sSum[w][tid];
        }
        wsMin[(size_t)(rb + tid) * CH + blockIdx.x] = mn;
        wsSum[(size_t)(rb + tid) * CH + blockIdx.x] = sm;
    }
}

__global__ __launch_bounds__(256) void lp_finalize(
    const float* __restrict__ wsMin,
    const float* __restrict__ wsSum,
    float* __restrict__ out)
{
    __shared__ float red[256];
    const int tid = threadIdx.x;
    float local = 0.0f;
    for (int r = tid; r < Bq; r += 256) {
        float pmin = __builtin_inff();
        float nsum = 0.0f;
        for (int c = 0; c < CH; ++c) {
            pmin  = fminf(pmin, wsMin[(size_t)r * CH + c]);
            nsum += wsSum[(size_t)r * CH + c];
        }
        local += -logf(pmin / (pmin + nsum + 1e-6f) + 1e-6f);
    }
    red[tid] = local;
    __syncthreads();
    for (int s = 128; s > 0; s >>= 1) {
        if (tid < s) red[tid] += red[tid + s];
        __syncthreads();
    }
    if (tid == 0) out[0] = red[0] / (float)Bq;
}

extern "C" void kernel_launch(void* const* d_in, const int* in_sizes, int n_in,
                              void* d_out, int out_size, void* d_ws, size_t ws_size,
                              hipStream_t stream) {
    const float* feats    = (const float*)d_in[0];   // [2048, 2048] f32
    const float* fs       = (const float*)d_in[1];   // [4096, 4, 2048] f32 -> [16384, 2048]
    const int*   labels   = (const int*)d_in[2];     // [2048]
    const int*   labels_s = (const int*)d_in[3];     // [16384]
    float*       out      = (float*)d_out;           // scalar f32

    float* wsMin = (float*)d_ws;                     // [Bq, CH]
    float* wsSum = wsMin + (size_t)Bq * CH;          // [Bq, CH]  (1 MB total scratch)

    dim3 grid(CH, Bq / 16);   // 64 x 128 blocks of 128 threads
    lp_gemm_reduce<<<grid, 128, 0, stream>>>(feats, fs, labels, labels_s, wsMin, wsSum);
    lp_finalize<<<1, 256, 0, stream>>>(wsMin, wsSum, out);
}